// SurvPathMamba_14620068675611
// MI455X (gfx1250) — compile-verified
//
// SurvPath-Mamba forward for MI455X (gfx1250), f32 WMMA everywhere.
// Bandwidth-bound (~500MB @ 23.3TB/s); all GEMMs via V_WMMA_F32_16X16X4_F32,
// double-buffered LDS pipeline, b128 global loads, 8B DS fragment reads.
#include <hip/hip_runtime.h>
#include <hip/hip_bf16.h>
#include <math.h>

#define D_MODEL    1024
#define D_INNER    2048
#define D_STATE    128
#define D_CONV     4
#define HEADDIM    64
#define NHEADS     32
#define D_IN_PROJ  4384
#define CONV_DIM   2304
#define NUM_PATCHES 32762
#define SUM_CHANNEL 32768
#define PROJ       256
#define SEQ_L      256

typedef __attribute__((ext_vector_type(2))) float v2f;
typedef __attribute__((ext_vector_type(8))) float v8f;

__device__ __forceinline__ float siluf(float x) { return x / (1.f + expf(-x)); }
__device__ __forceinline__ float geluf(float x) { return 0.5f * x * (1.f + erff(x * 0.70710678118654752f)); }

// ---------------------------------------------------------------------------
// Generic tiled f32 WMMA GEMM:  C = act(A@B + bias)
//   TRANS_A: A[m,k] read as Araw[k*lda + m]
//   TRANS_C: store C[n*ldc + m]
//   EPI: 0 none, 1 +bias, 2 +bias then exact GELU
// Requires K % 16 == 0 (true for all call sites: 256/1024/2048/32768).
// Block = 256 thr (8 waves), tile 64x64, BK=16; wave -> 16 rows x 32 cols.
// Double-buffered LDS: global->regs, WMMA(cur), regs->LDS(next), one barrier.
// ---------------------------------------------------------------------------
template <int TRANS_A, int TRANS_C, int EPI>
__global__ __launch_bounds__(256) void gemm_f32_wmma(
    const float* __restrict__ A, const float* __restrict__ B,
    const float* __restrict__ bias, float* __restrict__ C,
    int M, int N, int K, int lda, int ldb, int ldc)
{
  __shared__ float As[2][64][18];      // pitch 18: even (8B-aligned pairs), conflict-free
  __shared__ float Bs[2][8][64][2];    // [kpair][n][k&1]: fragment = one 8B read
  const int tid  = threadIdx.x;
  const int lane = tid & 31;
  const int wave = tid >> 5;
  const int wm   = wave >> 1;          // 0..3 (M tile)
  const int wn   = wave & 1;           // 0..1 (N tile pair)
  const int hi   = (lane >> 4) & 1;
  const int lr   = lane & 15;
  const int n0   = blockIdx.x * 64;
  const int m0   = blockIdx.y * 64;

  // per-thread staging coordinates (constant across K loop)
  const int rA  = tid >> 2;            // !TRANS_A: A row 0..63
  const int cA  = (tid & 3) << 2;      // !TRANS_A: A k-offset 0..12
  const int krA = tid >> 4;            // TRANS_A: k row 0..15
  const int mcA = (tid & 15) << 2;     // TRANS_A: m col 0..60
  const int rB  = tid >> 4;            // B k row 0..15
  const int cB  = (tid & 15) << 2;     // B n col 0..60

  const bool mokA  = (m0 + rA) < M;                       // !TRANS_A row guard
  const long rowA  = (long)(mokA ? (m0 + rA) : 0) * lda + cA;
  const bool mok4T = (m0 + mcA + 3) < M;                  // TRANS_A col guard
  const bool nok4  = (n0 + cB + 3) < N;                   // B col guard
  const long rowBk = (long)rB * ldb + n0 + cB;

  float ra[4], rb[4];

  auto fetch = [&](int k0) {
    if (TRANS_A) {
      long base = (long)(k0 + krA) * lda;                 // k in-bounds: K%16==0
      if (mok4T) {
        float4 t = *(const float4*)(A + base + m0 + mcA);
        ra[0] = t.x; ra[1] = t.y; ra[2] = t.z; ra[3] = t.w;
      } else {
#pragma unroll
        for (int i = 0; i < 4; ++i) {
          int gm = m0 + mcA + i;
          ra[i] = (gm < M) ? A[base + gm] : 0.f;
        }
      }
    } else {
      if (mokA) {
        float4 t = *(const float4*)(A + rowA + k0);
        ra[0] = t.x; ra[1] = t.y; ra[2] = t.z; ra[3] = t.w;
      } else {
        ra[0] = ra[1] = ra[2] = ra[3] = 0.f;
      }
    }
    if (nok4) {
      float4 t = *(const float4*)(B + (long)k0 * ldb + rowBk);
      rb[0] = t.x; rb[1] = t.y; rb[2] = t.z; rb[3] = t.w;
    } else {
      long base = (long)(k0 + rB) * ldb;
#pragma unroll
      for (int i = 0; i < 4; ++i) {
        int gn = n0 + cB + i;
        rb[i] = (gn < N) ? B[base + gn] : 0.f;
      }
    }
  };
  auto store = [&](int buf) {
    if (TRANS_A) {
#pragma unroll
      for (int i = 0; i < 4; ++i) As[buf][mcA + i][krA] = ra[i];
    } else {
#pragma unroll
      for (int i = 0; i < 4; ++i) As[buf][rA][cA + i] = ra[i];
    }
#pragma unroll
    for (int i = 0; i < 4; ++i) Bs[buf][rB >> 1][cB + i][rB & 1] = rb[i];
  };

  v8f acc0 = {0.f,0.f,0.f,0.f,0.f,0.f,0.f,0.f};
  v8f acc1 = {0.f,0.f,0.f,0.f,0.f,0.f,0.f,0.f};
  const int am  = wm * 16 + lr;
  const int bn0 = wn * 32 + lr;

  fetch(0);
  store(0);
  __syncthreads();
  int buf = 0;
  for (int k0 = 0; k0 < K; k0 += 16) {
    const bool more = (k0 + 16) < K;
    if (more) fetch(k0 + 16);          // global latency hidden behind WMMAs
#pragma unroll
    for (int kk = 0; kk < 16; kk += 4) {
      int ka = kk + 2 * hi;            // even: 8B-aligned fragment reads
      v2f a  = *(const v2f*)&As[buf][am][ka];
      v2f b0 = *(const v2f*)&Bs[buf][ka >> 1][bn0][0];
      v2f b1 = *(const v2f*)&Bs[buf][ka >> 1][bn0 + 16][0];
      acc0 = __builtin_amdgcn_wmma_f32_16x16x4_f32(false, a, false, b0, (short)0, acc0, false, false);
      acc1 = __builtin_amdgcn_wmma_f32_16x16x4_f32(false, a, false, b1, (short)0, acc1, false, false);
    }
    if (more) store(buf ^ 1);
    __syncthreads();
    buf ^= 1;
  }

#pragma unroll
  for (int t = 0; t < 2; ++t) {
    v8f acc = t ? acc1 : acc0;
    int n = n0 + wn * 32 + t * 16 + lr;
    if (n < N) {
      float bv = (EPI >= 1) ? bias[n] : 0.f;
#pragma unroll
      for (int r = 0; r < 8; ++r) {
        int m = m0 + wm * 16 + r + 8 * hi;   // C/D layout: VGPR r -> M=r / r+8
        if (m < M) {
          float v = acc[r] + bv;
          if (EPI == 2) v = geluf(v);
          if (TRANS_C) C[(long)n * ldc + m] = v;
          else         C[(long)m * ldc + n] = v;
        }
      }
    }
  }
}

// ---------------------------------------------------------------------------
// Exact per-feature quantiles (rank select) via 3-level radix histograms.
// ---------------------------------------------------------------------------
__device__ __forceinline__ unsigned f2key(float f) {
  unsigned u = __float_as_uint(f);
  return (u & 0x80000000u) ? ~u : (u | 0x80000000u);
}
__device__ __forceinline__ float key2f(unsigned k) {
  unsigned u = (k & 0x80000000u) ? (k & 0x7fffffffu) : ~k;
  return __uint_as_float(u);
}

__global__ __launch_bounds__(256) void quantiles_kernel(
    const float* __restrict__ X, int nrows, int ldx,
    float* __restrict__ med, float* __restrict__ scale)
{
  __shared__ unsigned h1[2048];     // level-1 (bits 31:21), shared by all ranks
  __shared__ unsigned hist[2048];
  __shared__ float vals[6];
  __shared__ unsigned s_bin, s_rank;
  const int j = blockIdx.x;
  const int tid = threadIdx.x;

  for (int i = tid; i < 2048; i += 256) h1[i] = 0;
  __syncthreads();
  for (int r = tid; r < nrows; r += 256)
    atomicAdd(&h1[f2key(X[(long)r * ldx + j]) >> 21], 1u);
  __syncthreads();

  const int ranks[6] = {8190, 8191, 16380, 16381, 24570, 24571};
  for (int q = 0; q < 6; ++q) {
    unsigned rank = (unsigned)ranks[q];
    if (tid == 0) {
      unsigned cum = 0, b = 0;
      for (; b < 2048; ++b) { unsigned c = h1[b]; if (cum + c > rank) break; cum += c; }
      s_bin = b; s_rank = rank - cum;
    }
    __syncthreads();
    unsigned prefix = s_bin << 21; rank = s_rank;

    for (int i = tid; i < 2048; i += 256) hist[i] = 0;
    __syncthreads();
    for (int r = tid; r < nrows; r += 256) {
      unsigned k = f2key(X[(long)r * ldx + j]);
      if ((k >> 21) == (prefix >> 21)) atomicAdd(&hist[(k >> 10) & 0x7FFu], 1u);
    }
    __syncthreads();
    if (tid == 0) {
      unsigned cum = 0, b = 0;
      for (; b < 2048; ++b) { unsigned c = hist[b]; if (cum + c > rank) break; cum += c; }
      s_bin = b; s_rank = rank - cum;
    }
    __syncthreads();
    prefix |= s_bin << 10; rank = s_rank;

    for (int i = tid; i < 2048; i += 256) hist[i] = 0;
    __syncthreads();
    for (int r = tid; r < nrows; r += 256) {
      unsigned k = f2key(X[(long)r * ldx + j]);
      if ((k >> 10) == (prefix >> 10)) atomicAdd(&hist[k & 0x3FFu], 1u);
    }
    __syncthreads();
    if (tid == 0) {
      unsigned cum = 0, b = 0;
      for (; b < 1024; ++b) { unsigned c = hist[b]; if (cum + c > rank) break; cum += c; }
      vals[q] = key2f(prefix | b);
    }
    __syncthreads();
  }
  if (tid == 0) {
    float q25 = vals[0] + 0.25f * (vals[1] - vals[0]);
    float m   = 0.5f  * (vals[2] + vals[3]);
    float q75 = vals[4] + 0.75f * (vals[5] - vals[4]);
    float s = q75 - q25;
    med[j] = m;
    scale[j] = (s == 0.f) ? 1.f : s;
  }
}

// ---------------------------------------------------------------------------
// Small elementwise / reduction kernels
// ---------------------------------------------------------------------------
__global__ __launch_bounds__(256) void snn_kernel(
    const float* __restrict__ x, int n,
    const float* __restrict__ w1, const float* __restrict__ b1,
    const float* __restrict__ w2, const float* __restrict__ b2,
    float* __restrict__ out)
{
  __shared__ float h1[256];
  int j = threadIdx.x;
  float s = b1[j];
  for (int i = 0; i < n; ++i) s += x[i] * w1[i * 256 + j];
  h1[j] = (s > 0.f) ? s : expm1f(s);          // ELU
  __syncthreads();
  float s2 = b2[j];
  for (int i = 0; i < 256; ++i) s2 += h1[i] * w2[i * 256 + j];
  out[j] = (s2 > 0.f) ? s2 : expm1f(s2);
}

__global__ void build_tokens_kernel(
    const float* __restrict__ wsi_raw, const float* __restrict__ h_omic,
    const float* __restrict__ med, const float* __restrict__ scale,
    float* __restrict__ tokens)
{
  int idx = blockIdx.x * 256 + threadIdx.x;   // SUM_CHANNEL*PROJ
  int c = idx >> 8, l = idx & 255;
  float v = (c < 6) ? h_omic[c * 256 + l]
                    : (wsi_raw[(long)(c - 6) * 256 + l] - med[l]) / scale[l];
  tokens[idx] = v;
}

__global__ void flip_rows_kernel(const float* __restrict__ in, float* __restrict__ out)
{
  int idx = blockIdx.x * 256 + threadIdx.x;   // SEQ_L*D_MODEL
  int l = idx >> 10, d = idx & 1023;
  out[idx] = in[(SEQ_L - 1 - l) * D_MODEL + d];
}

__global__ void dt_kernel(const float* __restrict__ zx, const float* __restrict__ dt_bias,
                          float* __restrict__ dt)
{
  int idx = blockIdx.x * 256 + threadIdx.x;   // SEQ_L*NHEADS
  if (idx >= SEQ_L * NHEADS) return;
  int l = idx >> 5, h = idx & 31;
  float v = zx[l * D_IN_PROJ + (D_INNER + CONV_DIM) + h] + dt_bias[h];
  dt[idx] = (v > 20.f) ? v : log1pf(expf(v)); // softplus
}

__global__ void conv_silu_kernel(const float* __restrict__ zx,
                                 const float* __restrict__ cw, const float* __restrict__ cb,
                                 float* __restrict__ out)
{
  int idx = blockIdx.x * 256 + threadIdx.x;   // SEQ_L*CONV_DIM
  int l = idx / CONV_DIM, ch = idx % CONV_DIM;
  float acc = cb[ch];
#pragma unroll
  for (int k = 0; k < D_CONV; ++k) {
    int ls = l - (D_CONV - 1) + k;
    if (ls >= 0) acc += zx[ls * D_IN_PROJ + D_INNER + ch] * cw[ch * D_CONV + k];
  }
  out[idx] = siluf(acc);
}

// One workgroup per head; 32 state elements per lane in registers.
__global__ __launch_bounds__(256) void ssm_kernel(
    const float* __restrict__ xbc, const float* __restrict__ dt,
    const float* __restrict__ A_log, const float* __restrict__ Dv,
    float* __restrict__ y)
{
  __shared__ float sB[D_STATE];
  __shared__ float sC[D_STATE];
  const int head = blockIdx.x;
  const int tid  = threadIdx.x;
  const int p    = tid >> 2;          // 0..63
  const int nq   = tid & 3;           // n-chunk
  const int nb   = nq * 32;
  const float Aneg = -expf(A_log[head]);
  const float Dh = Dv[head];
  float h[32];
#pragma unroll
  for (int i = 0; i < 32; ++i) h[i] = 0.f;

  for (int t = 0; t < SEQ_L; ++t) {
    __syncthreads();
    if (tid < 128) sB[tid] = xbc[t * CONV_DIM + D_INNER + tid];
    else           sC[tid - 128] = xbc[t * CONV_DIM + D_INNER + D_STATE + (tid - 128)];
    __syncthreads();
    float dtv = dt[t * NHEADS + head];
    float dAv = expf(dtv * Aneg);
    float xv  = xbc[t * CONV_DIM + head * HEADDIM + p];
    float xdt = xv * dtv;
    float part = 0.f;
#pragma unroll
    for (int i = 0; i < 32; ++i) {
      float hv = h[i] * dAv + xdt * sB[nb + i];
      h[i] = hv;
      part += hv * sC[nb + i];
    }
    part += __shfl_xor(part, 1, 32);
    part += __shfl_xor(part, 2, 32);
    if (nq == 0) y[t * D_INNER + head * HEADDIM + p] = part + xv * Dh;
  }
}

__global__ __launch_bounds__(256) void gate_rmsnorm_kernel(
    const float* __restrict__ y, const float* __restrict__ zx,
    const float* __restrict__ nw, float* __restrict__ out)
{
  __shared__ float red[256];
  int l = blockIdx.x, tid = threadIdx.x;
  float g[8]; float ss = 0.f;
#pragma unroll
  for (int i = 0; i < 8; ++i) {
    int ch = i * 256 + tid;
    float v = y[l * D_INNER + ch] * siluf(zx[l * D_IN_PROJ + ch]);
    g[i] = v; ss += v * v;
  }
  red[tid] = ss; __syncthreads();
  for (int s = 128; s > 0; s >>= 1) { if (tid < s) red[tid] += red[tid + s]; __syncthreads(); }
  float sc = rsqrtf(red[0] / (float)D_INNER + 1e-5f);
#pragma unroll
  for (int i = 0; i < 8; ++i) {
    int ch = i * 256 + tid;
    out[l * D_INNER + ch] = g[i] * sc * nw[ch];
  }
}

__global__ void combine_kernel(const float* __restrict__ yf, const float* __restrict__ yb,
                               float* __restrict__ out)
{
  int idx = blockIdx.x * 256 + threadIdx.x;   // SEQ_L*D_MODEL
  int l = idx >> 10, d = idx & 1023;
  out[idx] = yf[idx] + yb[(SEQ_L - 1 - l) * D_MODEL + d];
}

__global__ __launch_bounds__(256) void layernorm_kernel(
    const float* __restrict__ in, const float* __restrict__ g,
    const float* __restrict__ b, float* __restrict__ out)
{
  __shared__ float red[256];
  long base = (long)blockIdx.x * PROJ;
  int tid = threadIdx.x;
  float v = in[base + tid];
  red[tid] = v; __syncthreads();
  for (int s = 128; s > 0; s >>= 1) { if (tid < s) red[tid] += red[tid + s]; __syncthreads(); }
  float mu = red[0] / (float)PROJ;
  __syncthreads();
  float d = v - mu;
  red[tid] = d * d; __syncthreads();
  for (int s = 128; s > 0; s >>= 1) { if (tid < s) red[tid] += red[tid + s]; __syncthreads(); }
  out[base + tid] = d * rsqrtf(red[0] / (float)PROJ + 1e-5f) * g[tid] + b[tid];
}

__global__ __launch_bounds__(256) void pool_kernel(const float* __restrict__ mm,
                                                   float* __restrict__ emb)
{
  __shared__ float rp[256], rw[256];
  int j = blockIdx.x, tid = threadIdx.x;
  float sp = 0.f, sw = 0.f;
  for (int r = tid; r < SUM_CHANNEL; r += 256) {
    float v = mm[(long)r * PROJ + j];
    if (r < 6) sp += v; else sw += v;
  }
  rp[tid] = sp; rw[tid] = sw; __syncthreads();
  for (int s = 128; s > 0; s >>= 1) {
    if (tid < s) { rp[tid] += rp[tid + s]; rw[tid] += rw[tid + s]; }
    __syncthreads();
  }
  if (tid == 0) { emb[j] = rp[0] / 6.f; emb[PROJ + j] = rw[0] / (float)NUM_PATCHES; }
}

__global__ __launch_bounds__(64) void head_kernel(const float* __restrict__ emb,
    const float* __restrict__ w1, const float* __restrict__ b1,
    const float* __restrict__ w2, const float* __restrict__ b2,
    float* __restrict__ out)
{
  __shared__ float hid[64];
  int tid = threadIdx.x;
  float s = b1[tid];
  for (int k = 0; k < 512; ++k) s += emb[k] * w1[k * 64 + tid];
  hid[tid] = fmaxf(s, 0.f);
  __syncthreads();
  if (tid < 4) {
    float o = b2[tid];
    for (int k = 0; k < 64; ++k) o += hid[k] * w2[k * 4 + tid];
    out[tid] = o;
  }
}

// ---------------------------------------------------------------------------
static inline int cdiv(int a, int b) { return (a + b - 1) / b; }

extern "C" void kernel_launch(void* const* d_in, const int* in_sizes, int n_in,
                              void* d_out, int out_size, void* d_ws, size_t ws_size,
                              hipStream_t stream)
{
  (void)n_in; (void)out_size; (void)ws_size;
  // ---- input map: setup_inputs() dict order, params flattened depth-first ----
  const float* x_path = (const float*)d_in[0];
  const float* xo[6]; for (int i = 0; i < 6; ++i) xo[i] = (const float*)d_in[1 + i];
  int pi = 7;
  const float *sw1[6], *sb1[6], *sw2[6], *sb2[6];
  for (int i = 0; i < 6; ++i) {
    sw1[i] = (const float*)d_in[pi++]; sb1[i] = (const float*)d_in[pi++];
    sw2[i] = (const float*)d_in[pi++]; sb2[i] = (const float*)d_in[pi++];
  }
  const float* wsi_w = (const float*)d_in[pi++];
  pi++; // wsi_b: cancels exactly under RobustScaler (per-feature affine invariance)
  const float* fc_in  = (const float*)d_in[pi++];
  const float* fc_out = (const float*)d_in[pi++];
  const float *ip[2], *cw[2], *cb[2], *dtb[2], *alog[2], *Dp[2], *nw[2], *op[2];
  for (int d = 0; d < 2; ++d) {
    ip[d]   = (const float*)d_in[pi++]; cw[d]   = (const float*)d_in[pi++];
    cb[d]   = (const float*)d_in[pi++]; dtb[d]  = (const float*)d_in[pi++];
    alog[d] = (const float*)d_in[pi++]; Dp[d]   = (const float*)d_in[pi++];
    nw[d]   = (const float*)d_in[pi++]; op[d]   = (const float*)d_in[pi++];
  }
  const float* ff_ln_g = (const float*)d_in[pi++]; const float* ff_ln_b = (const float*)d_in[pi++];
  const float* ff_w1   = (const float*)d_in[pi++]; const float* ff_b1   = (const float*)d_in[pi++];
  const float* ff_w2   = (const float*)d_in[pi++]; const float* ff_b2   = (const float*)d_in[pi++];
  const float* ln_g    = (const float*)d_in[pi++]; const float* ln_b    = (const float*)d_in[pi++];
  const float* lg_w1   = (const float*)d_in[pi++]; const float* lg_b1   = (const float*)d_in[pi++];
  const float* lg_w2   = (const float*)d_in[pi++]; const float* lg_b2   = (const float*)d_in[pi++];

  // ---- workspace (bump allocator) ----
  char* base = (char*)d_ws; size_t off = 0;
  auto alloc = [&](size_t nfloat) -> float* {
    float* p = (float*)(base + off);
    off += ((nfloat * 4 + 255) / 256) * 256;
    return p;
  };
  float* big0 = alloc((size_t)SUM_CHANNEL * PROJ);   // wsi_raw -> ln1 -> ln2
  float* big1 = alloc((size_t)SUM_CHANNEL * PROJ);   // tokens  -> ff1
  float* big2 = alloc((size_t)SUM_CHANNEL * PROJ);   // fc_out^T -> ff2
  float* h_omic = alloc(6 * 256);
  float* med    = alloc(PROJ);
  float* scl    = alloc(PROJ);
  float* u      = alloc(SEQ_L * D_MODEL);
  float* uflip  = alloc(SEQ_L * D_MODEL);
  float* zx[2]   = { alloc(SEQ_L * D_IN_PROJ), alloc(SEQ_L * D_IN_PROJ) };
  float* dts[2]  = { alloc(SEQ_L * NHEADS),    alloc(SEQ_L * NHEADS) };
  float* xbc[2]  = { alloc(SEQ_L * CONV_DIM),  alloc(SEQ_L * CONV_DIM) };
  float* yssm[2] = { alloc(SEQ_L * D_INNER),   alloc(SEQ_L * D_INNER) };
  float* yg[2]   = { alloc(SEQ_L * D_INNER),   alloc(SEQ_L * D_INNER) };
  float* yo[2]   = { alloc(SEQ_L * D_MODEL),   alloc(SEQ_L * D_MODEL) };
  float* ysum    = alloc(SEQ_L * D_MODEL);
  float* emb     = alloc(2 * PROJ);

  dim3 blk(256);

  // 1) SNN towers -> h_omic (6 x 256)
  for (int i = 0; i < 6; ++i)
    snn_kernel<<<1, blk, 0, stream>>>(xo[i], in_sizes[1 + i], sw1[i], sb1[i], sw2[i], sb2[i],
                                      h_omic + i * 256);

  // 2) wsi_raw = x_path @ wsi_w  (32762x256, K=1024; bias/bound cancel)
  gemm_f32_wmma<0,0,0><<<dim3(cdiv(PROJ,64), cdiv(NUM_PATCHES,64)), blk, 0, stream>>>(
      x_path, wsi_w, nullptr, big0, NUM_PATCHES, PROJ, 1024, 1024, PROJ, PROJ);

  // 3) exact per-feature quantiles (RobustScaler)
  quantiles_kernel<<<PROJ, blk, 0, stream>>>(big0, NUM_PATCHES, PROJ, med, scl);

  // 4) tokens (32768 x 256)
  build_tokens_kernel<<<(SUM_CHANNEL * PROJ) / 256, blk, 0, stream>>>(big0, h_omic, med, scl, big1);

  // 5) u = tokens^T @ fc_in  (256 x 1024, K=32768) ; TRANS_A
  gemm_f32_wmma<1,0,0><<<dim3(cdiv(D_MODEL,64), cdiv(SEQ_L,64)), blk, 0, stream>>>(
      big1, fc_in, nullptr, u, SEQ_L, D_MODEL, SUM_CHANNEL, PROJ, D_MODEL, D_MODEL);

  // 6) flipped sequence for the backward direction
  flip_rows_kernel<<<(SEQ_L * D_MODEL) / 256, blk, 0, stream>>>(u, uflip);

  // 7) two Mamba2 directions
  for (int d = 0; d < 2; ++d) {
    const float* uin = d ? uflip : u;
    gemm_f32_wmma<0,0,0><<<dim3(cdiv(D_IN_PROJ,64), cdiv(SEQ_L,64)), blk, 0, stream>>>(
        uin, ip[d], nullptr, zx[d], SEQ_L, D_IN_PROJ, D_MODEL, D_MODEL, D_IN_PROJ, D_IN_PROJ);
    dt_kernel<<<cdiv(SEQ_L * NHEADS, 256), blk, 0, stream>>>(zx[d], dtb[d], dts[d]);
    conv_silu_kernel<<<(SEQ_L * CONV_DIM) / 256, blk, 0, stream>>>(zx[d], cw[d], cb[d], xbc[d]);
    ssm_kernel<<<NHEADS, blk, 0, stream>>>(xbc[d], dts[d], alog[d], Dp[d], yssm[d]);
    gate_rmsnorm_kernel<<<SEQ_L, blk, 0, stream>>>(yssm[d], zx[d], nw[d], yg[d]);
    gemm_f32_wmma<0,0,0><<<dim3(cdiv(D_MODEL,64), cdiv(SEQ_L,64)), blk, 0, stream>>>(
        yg[d], op[d], nullptr, yo[d], SEQ_L, D_MODEL, D_INNER, D_INNER, D_MODEL, D_MODEL);
  }
  combine_kernel<<<(SEQ_L * D_MODEL) / 256, blk, 0, stream>>>(yo[0], yo[1], ysum);

  // 8) mm^T = (ysum @ fc_out)^T stored directly as (32768 x 256) via TRANS_C
  gemm_f32_wmma<0,1,0><<<dim3(cdiv(SUM_CHANNEL,64), cdiv(SEQ_L,64)), blk, 0, stream>>>(
      ysum, fc_out, nullptr, big2, SEQ_L, SUM_CHANNEL, D_MODEL, D_MODEL, SUM_CHANNEL, PROJ);

  // 9) LN -> FF(gelu) -> FF -> LN
  layernorm_kernel<<<SUM_CHANNEL, blk, 0, stream>>>(big2, ff_ln_g, ff_ln_b, big0);
  gemm_f32_wmma<0,0,2><<<dim3(cdiv(PROJ,64), cdiv(SUM_CHANNEL,64)), blk, 0, stream>>>(
      big0, ff_w1, ff_b1, big1, SUM_CHANNEL, PROJ, PROJ, PROJ, PROJ, PROJ);
  gemm_f32_wmma<0,0,1><<<dim3(cdiv(PROJ,64), cdiv(SUM_CHANNEL,64)), blk, 0, stream>>>(
      big1, ff_w2, ff_b2, big2, SUM_CHANNEL, PROJ, PROJ, PROJ, PROJ, PROJ);
  layernorm_kernel<<<SUM_CHANNEL, blk, 0, stream>>>(big2, ln_g, ln_b, big0);

  // 10) pooled embedding + classifier head
  pool_kernel<<<PROJ, blk, 0, stream>>>(big0, emb);
  head_kernel<<<1, dim3(64), 0, stream>>>(emb, lg_w1, lg_b1, lg_w2, lg_b2, (float*)d_out);
}